// AggLayer_65730179498091
// MI455X (gfx1250) — compile-verified
//
#include <hip/hip_runtime.h>
#include <math.h>

typedef float v2f __attribute__((ext_vector_type(2)));
typedef float v8f __attribute__((ext_vector_type(8)));

// ---------------- utility kernels ----------------

__global__ void zero_f32(float* __restrict__ p, int n) {
  int i = blockIdx.x * blockDim.x + threadIdx.x;
  if (i < n) p[i] = 0.0f;
}

// deg[dst] += |w|
__global__ void deg_kernel(const int* __restrict__ dst, const float* __restrict__ w,
                           float* __restrict__ deg, int E) {
  int e = blockIdx.x * blockDim.x + threadIdx.x;
  if (e < E) atomicAdd(&deg[dst[e]], fabsf(w[e]));
}

// norm[e] = dis[src] * |w| * dis[dst],  dis = deg>0 ? rsqrt(deg) : 0
__global__ void norm_kernel(const int* __restrict__ src, const int* __restrict__ dst,
                            const float* __restrict__ w, const float* __restrict__ deg,
                            float* __restrict__ norm, int E) {
  int e = blockIdx.x * blockDim.x + threadIdx.x;
  if (e < E) {
    float ds = deg[src[e]];
    float dd = deg[dst[e]];
    float is = ds > 0.0f ? rsqrtf(ds) : 0.0f;
    float id = dd > 0.0f ? rsqrtf(dd) : 0.0f;
    norm[e] = is * fabsf(w[e]) * id;
  }
}

// feat[n*8+0] = x[n]  (channels >= din are never read)
__global__ void init_feat(const float* __restrict__ x, float* __restrict__ feat, int n) {
  int i = blockIdx.x * blockDim.x + threadIdx.x;
  if (i < n) feat[i * 8] = x[i];
}

// hout[dst*8+f] += norm[e] * hin[src*8+f], f < dim
__global__ void propagate(const int* __restrict__ src, const int* __restrict__ dst,
                          const float* __restrict__ norm,
                          const float* __restrict__ hin, float* __restrict__ hout,
                          int E, int dim) {
  int idx = blockIdx.x * blockDim.x + threadIdx.x;
  int total = E * dim;
  if (idx >= total) return;
  int e, f;
  if (dim == 8) { e = idx >> 3; f = idx & 7; }
  else          { e = idx;      f = 0; }
  float v = norm[e];
  float val = v * hin[src[e] * 8 + f];
  if (val != 0.0f) atomicAdd(&hout[dst[e] * 8 + f], val);
}

// ---------------- WMMA node-tile matmul: out += h @ W ----------------
// One wave per 16-node tile. A = 16 nodes x din (K padded to 8 with two
// 16x16x4 f32 WMMA steps), B = din x dout weights (cols padded to 16),
// C/D = 16 nodes x 16 channels f32 accumulator (chan >= dout masked).
__global__ void wmma_acc(const float* __restrict__ h, const float* __restrict__ W,
                         float* __restrict__ out, int nTiles, int din, int dout) {
  int wave = threadIdx.x >> 5;
  int lane = threadIdx.x & 31;
  int tile = blockIdx.x * (blockDim.x >> 5) + wave;
  if (tile >= nTiles) return;          // wave-uniform: EXEC stays all-ones at WMMA

  int half = lane >> 4;                // 0: lanes 0-15, 1: lanes 16-31
  int m    = lane & 15;                // node-in-tile for A, column for B/C
  int node = tile * 16 + m;
  int col  = m;

  // A (16x4 f32): VGPR0 holds K = 2*half, VGPR1 holds K = 2*half+1
  int k0 = 2 * half;
  v2f a0, a1;
  a0.x = (k0     < din) ? h[node * 8 + k0    ] : 0.0f;
  a0.y = (k0 + 1 < din) ? h[node * 8 + k0 + 1] : 0.0f;
  int k1 = k0 + 4;
  a1.x = (k1     < din) ? h[node * 8 + k1    ] : 0.0f;
  a1.y = (k1 + 1 < din) ? h[node * 8 + k1 + 1] : 0.0f;

  // B (4x16 f32): VGPR0 holds rows {0|2}, VGPR1 rows {1|3} (mirrors A striping)
  int r0 = 2 * half;
  v2f b0, b1;
  b0.x = (r0     < din && col < dout) ? W[(r0    ) * dout + col] : 0.0f;
  b0.y = (r0 + 1 < din && col < dout) ? W[(r0 + 1) * dout + col] : 0.0f;
  int r1 = r0 + 4;
  b1.x = (r1     < din && col < dout) ? W[(r1    ) * dout + col] : 0.0f;
  b1.y = (r1 + 1 < din && col < dout) ? W[(r1 + 1) * dout + col] : 0.0f;

  // C (16x16 f32): VGPR j -> node row j (+8 for upper half), channel = lane&15
  v8f c;
#pragma unroll
  for (int j = 0; j < 8; ++j) {
    int nd = tile * 16 + j + half * 8;
    c[j] = (col < dout) ? out[nd * 8 + col] : 0.0f;
  }

  c = __builtin_amdgcn_wmma_f32_16x16x4_f32(false, a0, false, b0, (short)0, c, false, false);
  c = __builtin_amdgcn_wmma_f32_16x16x4_f32(false, a1, false, b1, (short)0, c, false, false);

#pragma unroll
  for (int j = 0; j < 8; ++j) {
    int nd = tile * 16 + j + half * 8;
    if (col < dout) out[nd * 8 + col] = c[j];
  }
}

// feat = relu(acc + b) over channels < dout
__global__ void bias_relu(const float* __restrict__ acc, const float* __restrict__ b,
                          float* __restrict__ feat, int n, int dout) {
  int i = blockIdx.x * blockDim.x + threadIdx.x;
  if (i >= n * dout) return;
  int node = i / dout, c = i - node * dout;
  float v = acc[node * 8 + c] + b[c];
  feat[node * 8 + c] = fmaxf(v, 0.0f);
}

// ---------------- layernorm stats + normalize ----------------

__global__ void prep_kernel(float* __restrict__ stats, unsigned* __restrict__ sel,
                            unsigned* __restrict__ hist, const int* __restrict__ kptr) {
  int t = threadIdx.x;
  if (t < 256) hist[t] = 0u;
  if (t == 0) {
    stats[0] = 0.0f; stats[1] = 0.0f;
    sel[0] = 0u;                       // prefix
    sel[1] = (unsigned)kptr[0];        // remaining = k
    sel[2] = 0u;                       // tie counter
  }
}

__global__ void reduce_stats(const float* __restrict__ feat, float* __restrict__ stats, int n) {
  __shared__ float ssum[256];
  __shared__ float ssq[256];
  int t = threadIdx.x;
  int i = blockIdx.x * blockDim.x + t;
  float v = (i < n) ? feat[i * 8] : 0.0f;
  ssum[t] = v;
  ssq[t]  = v * v;
  __syncthreads();
  for (int s = 128; s > 0; s >>= 1) {
    if (t < s) { ssum[t] += ssum[t + s]; ssq[t] += ssq[t + s]; }
    __syncthreads();
  }
  if (t == 0) {
    atomicAdd(&stats[0], ssum[0]);
    atomicAdd(&stats[1], ssq[0]);
  }
}

__global__ void stats_final(float* __restrict__ stats, int n) {
  if (threadIdx.x == 0 && blockIdx.x == 0) {
    float inv_n = 1.0f / (float)n;
    float mean = stats[0] * inv_n;
    float var  = stats[1] * inv_n - mean * mean;
    stats[2] = mean;
    stats[3] = rsqrtf(var + 1e-5f);
  }
}

// out[2n] = (x-mean)*inv ; out[2n+1] = 0 ; keys = order-preserving uint of xnorm
__global__ void normalize_kernel(const float* __restrict__ feat, const float* __restrict__ stats,
                                 float* __restrict__ outp, unsigned* __restrict__ keys, int n) {
  int i = blockIdx.x * blockDim.x + threadIdx.x;
  if (i < n) {
    float xn = (feat[i * 8] - stats[2]) * stats[3];
    outp[2 * i]     = xn;
    outp[2 * i + 1] = 0.0f;
    unsigned u = __float_as_uint(xn);
    u = (u & 0x80000000u) ? ~u : (u | 0x80000000u);
    keys[i] = u;
  }
}

// ---------------- radix-select top-k (descending) ----------------

__global__ void hist_pass(const unsigned* __restrict__ keys, const unsigned* __restrict__ sel,
                          unsigned* __restrict__ hist, int n, int shift) {
  __shared__ unsigned sh[256];
  int t = threadIdx.x;
  if (t < 256) sh[t] = 0u;
  __syncthreads();
  int i = blockIdx.x * blockDim.x + t;
  if (i < n) {
    unsigned u = keys[i];
    unsigned maskHigh = (shift == 24) ? 0u : (0xFFFFFFFFu << (shift + 8));
    if ((u & maskHigh) == sel[0]) atomicAdd(&sh[(u >> shift) & 255u], 1u);
  }
  __syncthreads();
  if (t < 256 && sh[t]) atomicAdd(&hist[t], sh[t]);
}

__global__ void scan_pass(unsigned* __restrict__ sel, unsigned* __restrict__ hist, int shift) {
  int t = threadIdx.x;
  if (t == 0) {
    unsigned rem = sel[1], cum = 0u, pref = sel[0];
    for (int b = 255; b >= 0; --b) {
      unsigned c = hist[b];
      if (cum + c >= rem) {
        sel[0] = pref | ((unsigned)b << shift);
        sel[1] = rem - cum;
        break;
      }
      cum += c;
    }
  }
  __syncthreads();
  if (t < 256) hist[t] = 0u;   // reset for next pass
}

__global__ void mark_topk(const unsigned* __restrict__ keys, unsigned* __restrict__ sel,
                          float* __restrict__ outp, int n) {
  int i = blockIdx.x * blockDim.x + threadIdx.x;
  if (i < n) {
    unsigned u = keys[i];
    unsigned p = sel[0];
    if (u > p) {
      outp[2 * i + 1] = 1.0f;
    } else if (u == p) {
      unsigned pos = atomicAdd(&sel[2], 1u);
      if (pos < sel[1]) outp[2 * i + 1] = 1.0f;
    }
  }
}

// ---------------- host orchestration ----------------

extern "C" void kernel_launch(void* const* d_in, const int* in_sizes, int n_in,
                              void* d_out, int out_size, void* d_ws, size_t ws_size,
                              hipStream_t stream) {
  const float* x   = (const float*)d_in[0];
  const int*   ei  = (const int*)d_in[1];
  const float* ea  = (const float*)d_in[2];
  const int*   kpt = (const int*)d_in[3];
  const float* W1  = (const float*)d_in[4];
  const float* b1  = (const float*)d_in[5];
  const float* W2  = (const float*)d_in[6];
  const float* b2  = (const float*)d_in[7];
  const float* W3  = (const float*)d_in[8];
  const float* b3  = (const float*)d_in[9];
  float* out = (float*)d_out;

  const int N = in_sizes[0];
  const int E = in_sizes[2];
  const int* src = ei;
  const int* dst = ei + E;

  // workspace carve-out (256B aligned slices)
  char* ws = (char*)d_ws;
  size_t off = 0;
  auto alloc = [&](size_t bytes) -> void* {
    void* p = ws + off;
    off = (off + bytes + 255) & ~(size_t)255;
    return p;
  };
  float*    norm  = (float*)alloc((size_t)E * 4);
  float*    deg   = (float*)alloc((size_t)N * 4);
  float*    feat  = (float*)alloc((size_t)N * 32);
  float*    h     = (float*)alloc((size_t)N * 32);
  float*    h2    = (float*)alloc((size_t)N * 32);
  float*    acc   = (float*)alloc((size_t)N * 32);
  unsigned* keys  = (unsigned*)alloc((size_t)N * 4);
  float*    stats = (float*)alloc(64);
  unsigned* sel   = (unsigned*)alloc(64);
  unsigned* hist  = (unsigned*)alloc(1024);

  const int TB = 256;
  auto cdiv = [](int a, int b) { return (a + b - 1) / b; };

  // 1) GCN normalization
  zero_f32   <<<cdiv(N, TB), TB, 0, stream>>>(deg, N);
  deg_kernel <<<cdiv(E, TB), TB, 0, stream>>>(dst, ea, deg, E);
  norm_kernel<<<cdiv(E, TB), TB, 0, stream>>>(src, dst, ea, deg, norm, E);

  // 2) three TAGConv layers (WMMA for all h @ W[i] products)
  init_feat<<<cdiv(N, TB), TB, 0, stream>>>(x, feat, N);
  const int nTiles = cdiv(N, 16);
  const int wgrid  = cdiv(nTiles, TB / 32);

  struct Layer { const float* W; const float* b; int din, dout; };
  Layer layers[3] = { {W1, b1, 1, 8}, {W2, b2, 8, 8}, {W3, b3, 8, 1} };

  for (int L = 0; L < 3; ++L) {
    int din = layers[L].din, dout = layers[L].dout;
    const float* W = layers[L].W;
    zero_f32<<<cdiv(N * 8, TB), TB, 0, stream>>>(acc, N * 8);
    wmma_acc<<<wgrid, TB, 0, stream>>>(feat, W, acc, nTiles, din, dout);   // hop 0
    const float* cur = feat;
    float* bufs[2] = { h, h2 };
    for (int i = 1; i <= 3; ++i) {                                         // hops 1..K
      float* nxt = bufs[(i - 1) & 1];
      zero_f32<<<cdiv(N * 8, TB), TB, 0, stream>>>(nxt, N * 8);
      int total = E * din;
      propagate<<<cdiv(total, TB), TB, 0, stream>>>(src, dst, norm, cur, nxt, E, din);
      wmma_acc<<<wgrid, TB, 0, stream>>>(nxt, W + (size_t)i * din * dout, acc, nTiles, din, dout);
      cur = nxt;
    }
    bias_relu<<<cdiv(N * dout, TB), TB, 0, stream>>>(acc, layers[L].b, feat, N, dout);
  }

  // 3) layernorm over all N values + sortable keys
  prep_kernel     <<<1, 256, 0, stream>>>(stats, sel, hist, kpt);
  reduce_stats    <<<cdiv(N, TB), TB, 0, stream>>>(feat, stats, N);
  stats_final     <<<1, 1, 0, stream>>>(stats, N);
  normalize_kernel<<<cdiv(N, TB), TB, 0, stream>>>(feat, stats, out, keys, N);

  // 4) radix-select k-th largest, then mark top-k
  for (int p = 0; p < 4; ++p) {
    int shift = 24 - 8 * p;
    hist_pass<<<cdiv(N, TB), TB, 0, stream>>>(keys, sel, hist, N, shift);
    scan_pass<<<1, 256, 0, stream>>>(sel, hist, shift);
  }
  mark_topk<<<cdiv(N, TB), TB, 0, stream>>>(keys, sel, out, N);
}